// NeighborhoodCrossAttentionModule_82231443849832
// MI455X (gfx1250) — compile-verified
//
#include <hip/hip_runtime.h>
#include <hip/hip_bf16.h>
#include <cmath>

// ---------------------------------------------------------------------------
// NeighborhoodCrossAttention 3D  (B=1, C=128, D=16, H=48, W=48, heads=4, 3x3x3)
// MI455X / gfx1250: wave32, V_WMMA_F32_16X16X4_F32 for the dense GEMMs.
// ---------------------------------------------------------------------------

typedef __attribute__((ext_vector_type(2))) float v2f;
typedef __attribute__((ext_vector_type(8))) float v8f;

constexpr int CDIM  = 128;
constexpr int DD    = 16;
constexpr int HH    = 48;
constexpr int WW    = 48;
constexpr int NVOX  = DD * HH * WW;          // 36864
constexpr int HEADS = 4;
constexpr int HC    = CDIM / HEADS;          // 32 = head dim = wave32 lanes
constexpr float QSCALE = 0.17677669529663687f; // 32^-0.5

// workspace layout (floats)
constexpr size_t Q_OFF = 0;
constexpr size_t K_OFF = (size_t)HEADS * NVOX * HC;          // 4,718,592
constexpr size_t V_OFF = 2 * K_OFF;
constexpr size_t A_OFF = 3 * K_OFF;                          // attn out [n][128]

__device__ __forceinline__ v8f wmma_f32_16x16x4(v2f a, v2f b, v8f c) {
  // (neg_a, A, neg_b, B, c_mod, C, reuse_a, reuse_b)
  return __builtin_amdgcn_wmma_f32_16x16x4_f32(false, a, false, b, (short)0, c,
                                               false, false);
}

// ---------------------------------------------------------------------------
// Kernel 1: fused q/k/v projection.
//   qk[n,o]  = sum_c x[c,n]*Wqk[o,c] + bqk[o]   (q scaled by QSCALE)
//   vp[n,o]  = sum_c v[c,n]*Wv[o,c]  + bv[o]
// Block: 256 thr (8 waves), one 16-voxel tile, 24 channel tiles (3 per wave).
// ---------------------------------------------------------------------------
__global__ __launch_bounds__(256) void
k1_qkv_proj(const float* __restrict__ x, const float* __restrict__ vin,
            const float* __restrict__ Wqk, const float* __restrict__ bqk,
            const float* __restrict__ Wv,  const float* __restrict__ bv,
            float* __restrict__ ws) {
  __shared__ float ldsX[16 * 130];   // x  tile [m][c], pad 130 (8B-aligned pairs)
  __shared__ float ldsV[16 * 130];   // v  tile [m][c]

  const int n0   = blockIdx.x * 16;
  const int tid  = threadIdx.x;
  const int lane = tid & 31;
  const int wv   = tid >> 5;

  // stage x and v tiles: 2048 elements each, coalesced 64B rows
#pragma unroll
  for (int i = 0; i < 8; ++i) {
    int idx = tid + i * 256;
    int cc  = idx >> 4;
    int mm  = idx & 15;
    ldsX[mm * 130 + cc] = x  [(size_t)cc * NVOX + n0 + mm];
    ldsV[mm * 130 + cc] = vin[(size_t)cc * NVOX + n0 + mm];
  }
  __syncthreads();

  const int mrow  = lane & 15;        // A-fragment row (M)
  const int kksel = (lane >> 4) * 2;  // K pair selected by lane half
  const int col   = lane & 15;        // B/D column (N)

  float* qws = ws + Q_OFF;
  float* kws = ws + K_OFF;
  float* vws = ws + V_OFF;

#pragma unroll
  for (int tt = 0; tt < 3; ++tt) {
    const int t  = wv + tt * 8;       // 0..23 channel tiles
    const int o0 = t * 16;

    const float* ldsA;
    const float* wmat;
    int orow;
    if (o0 < 256) { ldsA = ldsX; wmat = Wqk; orow = o0; }       // q or k
    else          { ldsA = ldsV; wmat = Wv;  orow = o0 - 256; } // v

    const float* wrow = wmat + (size_t)(orow + col) * CDIM;

    v8f acc = {};
#pragma unroll
    for (int c0 = 0; c0 < CDIM; c0 += 4) {
      v2f a, b;
      a.x = ldsA[mrow * 130 + c0 + kksel];
      a.y = ldsA[mrow * 130 + c0 + kksel + 1];
      b.x = wrow[c0 + kksel];
      b.y = wrow[c0 + kksel + 1];
      acc = wmma_f32_16x16x4(a, b, acc);
    }

    // epilogue: bias (+scale for q), store head-major [h][n][32]
    float badd, mul;
    float* dst;
    int head, cc0;
    if (o0 < 128) {            // q
      badd = bqk[o0 + col]; mul = QSCALE;
      dst = qws; head = o0 >> 5; cc0 = o0 & 31;
    } else if (o0 < 256) {     // k
      badd = bqk[o0 + col]; mul = 1.0f;
      dst = kws; head = (o0 - 128) >> 5; cc0 = (o0 - 128) & 31;
    } else {                   // v
      badd = bv[o0 - 256 + col]; mul = 1.0f;
      dst = vws; head = (o0 - 256) >> 5; cc0 = (o0 - 256) & 31;
    }
    float* base = dst + (size_t)head * NVOX * HC;
#pragma unroll
    for (int r = 0; r < 8; ++r) {
      int m = r + ((lane >> 4) << 3);
      base[(size_t)(n0 + m) * HC + cc0 + col] = (acc[r] + badd) * mul;
    }
  }
}

// ---------------------------------------------------------------------------
// Kernel 2: 3x3x3 clamped-window neighborhood attention + RPB + softmax + PV.
// One wave per (head, voxel); lane = head channel (32 lanes).
// Output voxel-major: a_ws[n*128 + head*32 + lane]  (contiguous 128B stores)
// ---------------------------------------------------------------------------
__global__ __launch_bounds__(256) void
k2_natten(const float* __restrict__ ws_in, const float* __restrict__ rpb,
          float* __restrict__ aout) {
  const int wid  = blockIdx.x * 8 + (threadIdx.x >> 5);
  const int lane = threadIdx.x & 31;

  const int head = wid / NVOX;
  const int n    = wid - head * NVOX;
  const int d    = n / (HH * WW);
  const int rem  = n - d * HH * WW;
  const int hy   = rem / WW;
  const int wx   = rem - hy * WW;

  const int sd = min(max(d  - 1, 0), DD - 3);
  const int sh = min(max(hy - 1, 0), HH - 3);
  const int sw = min(max(wx - 1, 0), WW - 3);

  const float* qb = ws_in + Q_OFF + ((size_t)head * NVOX + n) * HC;
  const float* kb = ws_in + K_OFF + (size_t)head * NVOX * HC;
  const float* vb = ws_in + V_OFF + (size_t)head * NVOX * HC;
  const float* rb = rpb + head * 125;   // [5][5][5]

  const float qv = qb[lane];

  float logits[27];
  float mx = -1e30f;
#pragma unroll
  for (int a = 0; a < 3; ++a)
#pragma unroll
    for (int b = 0; b < 3; ++b)
#pragma unroll
      for (int c = 0; c < 3; ++c) {
        const int kn = ((sd + a) * HH + (sh + b)) * WW + (sw + c);
        float p = qv * kb[(size_t)kn * HC + lane];
        p += __shfl_xor(p, 1, 32);
        p += __shfl_xor(p, 2, 32);
        p += __shfl_xor(p, 4, 32);
        p += __shfl_xor(p, 8, 32);
        p += __shfl_xor(p, 16, 32);
        const float bias =
            rb[((sd + a - d + 2) * 5 + (sh + b - hy + 2)) * 5 + (sw + c - wx + 2)];
        const float lg = p + bias;
        logits[(a * 3 + b) * 3 + c] = lg;
        mx = fmaxf(mx, lg);
      }

  float denom = 0.0f, outv = 0.0f;
#pragma unroll
  for (int j = 0; j < 27; ++j) {
    const int a = j / 9, b = (j / 3) % 3, c = j % 3;
    const int kn = ((sd + a) * HH + (sh + b)) * WW + (sw + c);
    const float e = __expf(logits[j] - mx);
    denom += e;
    outv = fmaf(e, vb[(size_t)kn * HC + lane], outv);
  }
  aout[(size_t)n * CDIM + head * HC + lane] = outv / denom;
}

// ---------------------------------------------------------------------------
// Kernel 3: output projection  out[co,n] = sum_ci a[n,ci]*Wp[co,ci] + bp[co]
// A is voxel-major -> stage to LDS; per-wave LDS transpose for coalesced
// channel-major stores. Block: 16 voxels, 8 waves x one 16-channel tile.
// ---------------------------------------------------------------------------
__global__ __launch_bounds__(256) void
k3_out_proj(const float* __restrict__ ain, const float* __restrict__ Wp,
            const float* __restrict__ bp, float* __restrict__ out) {
  __shared__ float ldsA[16 * 130];       // a tile [m][ci]
  __shared__ float ldsT[8 * 16 * 17];    // per-wave transpose tile [co][m]

  const int n0   = blockIdx.x * 16;
  const int tid  = threadIdx.x;
  const int lane = tid & 31;
  const int wv   = tid >> 5;

#pragma unroll
  for (int i = 0; i < 8; ++i) {
    int idx = tid + i * 256;
    int mm  = idx >> 7;
    int ci  = idx & 127;
    ldsA[mm * 130 + ci] = ain[(size_t)(n0 + mm) * CDIM + ci];
  }
  __syncthreads();

  const int mrow  = lane & 15;
  const int kksel = (lane >> 4) * 2;
  const int col   = lane & 15;
  const int o0    = wv * 16;

  const float* wrow = Wp + (size_t)(o0 + col) * CDIM;

  v8f acc = {};
#pragma unroll
  for (int c0 = 0; c0 < CDIM; c0 += 4) {
    v2f a, b;
    a.x = ldsA[mrow * 130 + c0 + kksel];
    a.y = ldsA[mrow * 130 + c0 + kksel + 1];
    b.x = wrow[c0 + kksel];
    b.y = wrow[c0 + kksel + 1];
    acc = wmma_f32_16x16x4(a, b, acc);
  }

  const float badd = bp[o0 + col];
  float* tbase = ldsT + wv * (16 * 17);
#pragma unroll
  for (int r = 0; r < 8; ++r) {
    int m = r + ((lane >> 4) << 3);        // voxel row
    tbase[col * 17 + m] = acc[r] + badd;   // [co][m]
  }
  // wave-private tile: DS ops from the same wave are in-order via dscnt waits
#pragma unroll
  for (int r2 = 0; r2 < 8; ++r2) {
    int co = 2 * r2 + (lane >> 4);
    int nl = lane & 15;
    out[(size_t)(o0 + co) * NVOX + n0 + nl] = tbase[co * 17 + nl];
  }
}

// ---------------------------------------------------------------------------
extern "C" void kernel_launch(void* const* d_in, const int* in_sizes, int n_in,
                              void* d_out, int out_size, void* d_ws, size_t ws_size,
                              hipStream_t stream) {
  const float* x    = (const float*)d_in[0];
  const float* vin  = (const float*)d_in[1];
  const float* Wqk  = (const float*)d_in[2];
  const float* bqk  = (const float*)d_in[3];
  const float* Wv   = (const float*)d_in[4];
  const float* bv   = (const float*)d_in[5];
  const float* rpb  = (const float*)d_in[6];
  const float* Wp   = (const float*)d_in[7];
  const float* bp   = (const float*)d_in[8];
  float* out = (float*)d_out;
  float* ws  = (float*)d_ws;

  dim3 blk(256);
  // 1) fused qkv projection: 36864/16 = 2304 voxel tiles
  k1_qkv_proj<<<dim3(NVOX / 16), blk, 0, stream>>>(x, vin, Wqk, bqk, Wv, bv, ws);
  // 2) neighborhood attention: HEADS*NVOX waves / 8 waves per block
  k2_natten<<<dim3(HEADS * NVOX / 8), blk, 0, stream>>>(ws, rpb, ws + A_OFF);
  // 3) output projection
  k3_out_proj<<<dim3(NVOX / 16), blk, 0, stream>>>(ws + A_OFF, Wp, bp, out);
}